// SimpleGATLayer_19713899889532
// MI455X (gfx1250) — compile-verified
//
#include <hip/hip_runtime.h>

#define N_NODES 4096
#define C 128
#define HEADS 8
#define HDIM 16
#define MASK_WPR (N_NODES / 32)   // 128 mask words per row

typedef float v2f __attribute__((ext_vector_type(2)));
typedef float v8f __attribute__((ext_vector_type(8)));

__device__ __forceinline__ v8f wmma4(v2f a, v2f b, v8f c) {
    // D = A(16x4,f32) x B(4x16,f32) + C(16x16,f32)
    return __builtin_amdgcn_wmma_f32_16x16x4_f32(
        /*neg_a=*/false, a, /*neg_b=*/false, b,
        /*c_mod=*/(short)0, c, /*reuse_a=*/false, /*reuse_b=*/false);
}

// ---------------- mask build ----------------
__global__ void k_zero_u32(unsigned* __restrict__ p, int n) {
    int i = blockIdx.x * blockDim.x + threadIdx.x;
    if (i < n) p[i] = 0u;
}

__global__ void k_scatter_edges(const int* __restrict__ ei, int E,
                                unsigned* __restrict__ maskw) {
    int e = blockIdx.x * blockDim.x + threadIdx.x;
    if (e >= E) return;
    int s = ei[e];        // row (query n)
    int d = ei[E + e];    // col (key m)
    s = min(max(s, 0), N_NODES - 1);
    d = min(max(d, 0), N_NODES - 1);
    atomicOr(&maskw[s * MASK_WPR + (d >> 5)], 1u << (d & 31));
}

// ---------------- WMMA f32 GEMM: Out[M x 128] = A[M x 128] @ W[128 x 128] + bias ----------------
__global__ __launch_bounds__(32) void k_gemm(const float* __restrict__ A,
                                             const float* __restrict__ W,
                                             const float* __restrict__ bias,
                                             float* __restrict__ Out) {
    const int i16 = blockIdx.x * 16;   // row tile
    const int j16 = blockIdx.y * 16;   // col tile
    const int lane = threadIdx.x;
    const int half = lane >> 4;        // 0 or 1
    const int l15  = lane & 15;

    v8f acc = {};
#pragma unroll 4
    for (int kc = 0; kc < C; kc += 4) {
        const int kk = kc + half * 2;
        v2f a = *(const v2f*)(A + (size_t)(i16 + l15) * C + kk);   // A[row][kk..kk+1]
        v2f b;
        b.x = W[(size_t)(kk + 0) * C + j16 + l15];                 // B[k][col]
        b.y = W[(size_t)(kk + 1) * C + j16 + l15];
        acc = wmma4(a, b, acc);
    }
    const float bv = bias[j16 + l15];
#pragma unroll
    for (int r = 0; r < 8; ++r) {
        const int row = i16 + r + half * 8;
        Out[(size_t)row * C + j16 + l15] = acc[r] + bv;
    }
}

// ---------------- column mean of V (for fully-masked rows: softmax NaN -> 1/n) ----------------
__global__ void k_colmean(const float* __restrict__ V, float* __restrict__ meanV) {
    __shared__ float red[256];
    const int c = blockIdx.x;
    float s = 0.f;
    for (int n = threadIdx.x; n < N_NODES; n += 256) s += V[(size_t)n * C + c];
    red[threadIdx.x] = s;
    __syncthreads();
    for (int off = 128; off > 0; off >>= 1) {
        if (threadIdx.x < (unsigned)off) red[threadIdx.x] += red[threadIdx.x + off];
        __syncthreads();
    }
    if (threadIdx.x == 0) meanV[c] = red[0] * (1.0f / N_NODES);
}

// ---------------- fused masked attention: one wave per (16-row tile, head) ----------------
// Softmax without running-max (scores ~N(0,1/3): exp is overflow-safe by huge margin;
// mathematically identical result). Denominator accumulated per-lane, reduced once at end.
__global__ __launch_bounds__(32) void k_attn(const float* __restrict__ Q,
                                             const float* __restrict__ K,
                                             const float* __restrict__ V,
                                             const unsigned* __restrict__ maskw,
                                             const float* __restrict__ meanV,
                                             float* __restrict__ attn) {
    __shared__ float tile[16][17];   // P transpose staging (padded)
    const int n0 = blockIdx.x * 16;
    const int hd = blockIdx.y * HDIM;
    const int lane = threadIdx.x;
    const int half = lane >> 4;
    const int l15  = lane & 15;

    // Q tile in A-layout chunks (resident for whole m-loop)
    v2f qa[4];
#pragma unroll
    for (int c = 0; c < 4; ++c)
        qa[c] = *(const v2f*)(Q + (size_t)(n0 + l15) * C + hd + c * 4 + half * 2);

    v8f acc = {};
    float lsum[8];
#pragma unroll
    for (int r = 0; r < 8; ++r) lsum[r] = 0.f;

    for (int m0 = 0; m0 < N_NODES; m0 += 16) {
        const int wofs = m0 >> 5;
        const int bofs = m0 & 31;

        // mask bits for my 8 rows (16-bit field per row); wave-uniform empty-tile skip
        unsigned fl[8];
        unsigned orall = 0;
#pragma unroll
        for (int r = 0; r < 8; ++r) {
            const unsigned w = maskw[(size_t)(n0 + r + half * 8) * MASK_WPR + wofs];
            fl[r] = (w >> bofs) & 0xFFFFu;
            orall |= fl[r];
        }
        if (__ballot(orall != 0) == 0ull) continue;  // uniform branch: EXEC stays all-1s

        // ---- S = Q K^T for this 16x16 tile (K-dim 16 in 4 chunks of 4) ----
        v8f s = {};
#pragma unroll
        for (int c = 0; c < 4; ++c) {
            v2f kb = *(const v2f*)(K + (size_t)(m0 + l15) * C + hd + c * 4 + half * 2);
            s = wmma4(qa[c], kb, s);
        }

        // ---- masked exp; per-lane partial denominators ----
        float p[8];
#pragma unroll
        for (int r = 0; r < 8; ++r) {
            const bool on = (fl[r] >> l15) & 1u;
            const float pv = on ? __expf(s[r] * 0.25f) : 0.f;   // scale = 1/sqrt(16)
            lsum[r] += pv;
            p[r] = pv;
        }

        // ---- re-layout P (C-layout -> A-layout) through LDS ----
#pragma unroll
        for (int r = 0; r < 8; ++r) tile[r + half * 8][l15] = p[r];
        asm volatile("s_wait_dscnt 0" ::: "memory");

        // ---- acc += P @ V (K-dim 16 over m, 4 chunks of 4) ----
#pragma unroll
        for (int c = 0; c < 4; ++c) {
            v2f pa;
            pa.x = tile[l15][c * 4 + half * 2 + 0];
            pa.y = tile[l15][c * 4 + half * 2 + 1];
            v2f vb;
            vb.x = V[(size_t)(m0 + c * 4 + half * 2 + 0) * C + hd + l15];
            vb.y = V[(size_t)(m0 + c * 4 + half * 2 + 1) * C + hd + l15];
            acc = wmma4(pa, vb, acc);
        }
    }

    // ---- final row-sum reduction; O/l, or mean(V) column for fully-masked rows ----
#pragma unroll
    for (int r = 0; r < 8; ++r) {
        float rs = lsum[r];
        rs += __shfl_xor(rs, 1);
        rs += __shfl_xor(rs, 2);
        rs += __shfl_xor(rs, 4);
        rs += __shfl_xor(rs, 8);
        const int n = n0 + r + half * 8;
        float o;
        if (rs > 0.f) o = acc[r] / rs;
        else          o = meanV[hd + l15];
        attn[(size_t)n * C + hd + l15] = o;
    }
}

// ---------------- launcher ----------------
extern "C" void kernel_launch(void* const* d_in, const int* in_sizes, int n_in,
                              void* d_out, int out_size, void* d_ws, size_t ws_size,
                              hipStream_t stream) {
    const float* x  = (const float*)d_in[0];
    const int*   ei = (const int*)d_in[1];
    const float* Wq = (const float*)d_in[2];
    const float* bq = (const float*)d_in[3];
    const float* Wk = (const float*)d_in[4];
    const float* bk = (const float*)d_in[5];
    const float* Wv = (const float*)d_in[6];
    const float* bv = (const float*)d_in[7];
    const float* Wo = (const float*)d_in[8];
    const float* bo = (const float*)d_in[9];
    float* out = (float*)d_out;

    const size_t NC = (size_t)N_NODES * C;
    float* ws     = (float*)d_ws;
    float* Qb     = ws;
    float* Kb     = ws + NC;
    float* Vb     = ws + 2 * NC;
    float* attnB  = ws + 3 * NC;
    float* meanV  = ws + 4 * NC;                       // 128 floats
    unsigned* maskw = (unsigned*)(ws + 4 * NC + 128);  // 4096*128 words (2 MB)

    const int E = in_sizes[1] / 2;
    const int nMaskWords = N_NODES * MASK_WPR;

    k_zero_u32<<<(nMaskWords + 255) / 256, 256, 0, stream>>>(maskw, nMaskWords);
    k_scatter_edges<<<(E + 255) / 256, 256, 0, stream>>>(ei, E, maskw);

    dim3 gg(N_NODES / 16, C / 16);
    k_gemm<<<gg, 32, 0, stream>>>(x, Wq, bq, Qb);
    k_gemm<<<gg, 32, 0, stream>>>(x, Wk, bk, Kb);
    k_gemm<<<gg, 32, 0, stream>>>(x, Wv, bv, Vb);

    k_colmean<<<C, 256, 0, stream>>>(Vb, meanV);

    dim3 ga(N_NODES / 16, HEADS);
    k_attn<<<ga, 32, 0, stream>>>(Qb, Kb, Vb, maskw, meanV, attnB);

    k_gemm<<<gg, 32, 0, stream>>>(attnB, Wo, bo, out);
}